// QuaternionicGaussLinkingLoss_56126632624409
// MI455X (gfx1250) — compile-verified
//
#include <hip/hip_runtime.h>
#include <math.h>

typedef __attribute__((ext_vector_type(2))) float v2f;
typedef __attribute__((ext_vector_type(8))) float v8f;

#define BQ 4
#define LQ 2048
#define NSEG 2047          // L-1 segments
#define NPAD 2048          // padded to multiple of 16
#define NTILE 128          // NPAD / 16 tiles per dimension
#define TILES_PER_BATCH (NTILE * NTILE)        // 16384
#define WAVES_PER_BLOCK 8
#define BLOCKS_PER_BATCH (TILES_PER_BATCH / WAVES_PER_BLOCK)  // 2048
#define EPSF 1e-6f

// ---------------------------------------------------------------------------
// Phase 1: per-segment feature extraction.
// Feature row (16 floats, 64B) per segment:
//   A-role (path a): [ ua.x ua.y ua.z  da.x da.y da.z  0 0 | |ra|^2  1  ra.x ra.y ra.z  0 0 0 ]
//   B-role (path b): [ db.x db.y db.z  vb.x vb.y vb.z  0 0 |   1  |rb|^2 -2rb.x -2rb.y -2rb.z 0 0 0 ]
// where u = r x dr (for a), v = r x dr (for b).  Then
//   num[i,j]  = sum_{k=0..7}  Fa[i][k]   * Fb[j][k]    (triple product, rank-6)
//   dist2[i,j]= sum_{k=8..15} Fa[i][k]   * Fb[j][k]    (|ra|^2+|rb|^2-2 ra.rb)
// Segment index NSEG (padding) is written as all zeros -> contributes exactly 0.
// ---------------------------------------------------------------------------
__device__ inline void quat_imag(const float* __restrict__ q, float p[3]) {
    float w = q[0], x = q[1], y = q[2], z = q[3];
    float n = sqrtf(w * w + x * x + y * y + z * z) + 1e-12f;
    float inv = 1.0f / n;
    p[0] = x * inv; p[1] = y * inv; p[2] = z * inv;
}

__global__ void __launch_bounds__(256)
seg_features(const float* __restrict__ gq, const float* __restrict__ cq,
             float* __restrict__ Fa, float* __restrict__ Fb) {
    int idx = blockIdx.x * blockDim.x + threadIdx.x;   // [0, BQ*NPAD)
    if (idx >= BQ * NPAD) return;
    int b = idx / NPAD;
    int i = idx % NPAD;

    float fa[16], fb[16];
    if (i < NSEG) {
        float pa0[3], pa1[3], pb0[3], pb1[3];
        quat_imag(gq + (size_t)(b * LQ + i) * 4,     pa0);
        quat_imag(gq + (size_t)(b * LQ + i + 1) * 4, pa1);
        quat_imag(cq + (size_t)(b * LQ + i) * 4,     pb0);
        quat_imag(cq + (size_t)(b * LQ + i + 1) * 4, pb1);

        float da[3], ra[3], db[3], rb[3];
        for (int c = 0; c < 3; ++c) {
            da[c] = pa1[c] - pa0[c];  ra[c] = 0.5f * (pa1[c] + pa0[c]);
            db[c] = pb1[c] - pb0[c];  rb[c] = 0.5f * (pb1[c] + pb0[c]);
        }
        // u = ra x da ; v = rb x db
        float ua[3] = { ra[1]*da[2] - ra[2]*da[1],
                        ra[2]*da[0] - ra[0]*da[2],
                        ra[0]*da[1] - ra[1]*da[0] };
        float vb[3] = { rb[1]*db[2] - rb[2]*db[1],
                        rb[2]*db[0] - rb[0]*db[2],
                        rb[0]*db[1] - rb[1]*db[0] };
        float ra2 = ra[0]*ra[0] + ra[1]*ra[1] + ra[2]*ra[2];
        float rb2 = rb[0]*rb[0] + rb[1]*rb[1] + rb[2]*rb[2];

        fa[0]=ua[0]; fa[1]=ua[1]; fa[2]=ua[2]; fa[3]=da[0]; fa[4]=da[1]; fa[5]=da[2];
        fa[6]=0.f;   fa[7]=0.f;
        fa[8]=ra2;   fa[9]=1.f;   fa[10]=ra[0]; fa[11]=ra[1]; fa[12]=ra[2];
        fa[13]=0.f;  fa[14]=0.f;  fa[15]=0.f;

        fb[0]=db[0]; fb[1]=db[1]; fb[2]=db[2]; fb[3]=vb[0]; fb[4]=vb[1]; fb[5]=vb[2];
        fb[6]=0.f;   fb[7]=0.f;
        fb[8]=1.f;   fb[9]=rb2;   fb[10]=-2.f*rb[0]; fb[11]=-2.f*rb[1]; fb[12]=-2.f*rb[2];
        fb[13]=0.f;  fb[14]=0.f;  fb[15]=0.f;
    } else {
        #pragma unroll
        for (int k = 0; k < 16; ++k) { fa[k] = 0.f; fb[k] = 0.f; }
    }

    float4* oa = (float4*)(Fa + (size_t)idx * 16);
    float4* ob = (float4*)(Fb + (size_t)idx * 16);
    #pragma unroll
    for (int k = 0; k < 4; ++k) {
        oa[k] = make_float4(fa[4*k], fa[4*k+1], fa[4*k+2], fa[4*k+3]);
        ob[k] = make_float4(fb[4*k], fb[4*k+1], fb[4*k+2], fb[4*k+3]);
    }
}

// ---------------------------------------------------------------------------
// Phase 2: one wave32 per 16x16 tile of the (i,j) pair grid.
// A operand layout (f32 16x4): lanes 0-15 hold M=lane, K={0,1} in vgpr {0,1};
//                              lanes 16-31 hold M=lane-16, K={2,3}.
// B operand layout (f32 4x16): lanes 0-15 hold N=lane, K rows {0,1};
//                              lanes 16-31 hold N=lane-16, K rows {2,3}.
// -> lane (m = lane&15, ksel = lane>>4) loads float2 at feature offset ksel*2,
//    chunk1 at +4; dist features at +8 / +12.  Two chained K=4 WMMAs per GEMM.
// ---------------------------------------------------------------------------
__global__ void __launch_bounds__(256)
gauss_tiles(const float* __restrict__ Fa, const float* __restrict__ Fb,
            float* __restrict__ partials) {
    const int lane  = threadIdx.x & 31;
    const int wave  = threadIdx.x >> 5;
    const int tile  = blockIdx.x * WAVES_PER_BLOCK + wave;
    const int batch = tile / TILES_PER_BATCH;
    const int t     = tile % TILES_PER_BATCH;
    const int ti    = t >> 7;     // i-tile
    const int tj    = t & 127;    // j-tile

    const int m    = lane & 15;
    const int ksel = lane >> 4;

    const float* pa = Fa + ((size_t)(batch * NPAD + ti * 16 + m)) * 16 + ksel * 2;
    const float* pb = Fb + ((size_t)(batch * NPAD + tj * 16 + m)) * 16 + ksel * 2;

    v2f an0 = *(const v2f*)(pa + 0);    // numerator K 0..3
    v2f an1 = *(const v2f*)(pa + 4);    // numerator K 4..7
    v2f ad0 = *(const v2f*)(pa + 8);    // dist2     K 0..3
    v2f ad1 = *(const v2f*)(pa + 12);   // dist2     K 4..7
    v2f bn0 = *(const v2f*)(pb + 0);
    v2f bn1 = *(const v2f*)(pb + 4);
    v2f bd0 = *(const v2f*)(pb + 8);
    v2f bd1 = *(const v2f*)(pb + 12);

    v8f cn = {};
    v8f cd = {};
    cn = __builtin_amdgcn_wmma_f32_16x16x4_f32(false, an0, false, bn0, (short)0, cn, false, false);
    cn = __builtin_amdgcn_wmma_f32_16x16x4_f32(false, an1, false, bn1, (short)0, cn, false, false);
    cd = __builtin_amdgcn_wmma_f32_16x16x4_f32(false, ad0, false, bd0, (short)0, cd, false, false);
    cd = __builtin_amdgcn_wmma_f32_16x16x4_f32(false, ad1, false, bd1, (short)0, cd, false, false);

    // Per-pair epilogue: kernel = num / (sqrt(d2)+eps)^3
    float acc = 0.f;
    #pragma unroll
    for (int v = 0; v < 8; ++v) {
        float d2 = cd[v] < 0.f ? 0.f : cd[v];          // clamp cancellation noise
        float dist = __builtin_amdgcn_sqrtf(d2) + EPSF;
        float d3 = dist * dist * dist;
        acc += cn[v] * __builtin_amdgcn_rcpf(d3);
    }

    // wave32 butterfly reduction
    #pragma unroll
    for (int off = 16; off > 0; off >>= 1)
        acc += __shfl_xor(acc, off, 32);

    __shared__ float sp[WAVES_PER_BLOCK];
    if (lane == 0) sp[wave] = acc;
    __syncthreads();
    if (threadIdx.x == 0) {
        float s = 0.f;
        #pragma unroll
        for (int w = 0; w < WAVES_PER_BLOCK; ++w) s += sp[w];
        partials[blockIdx.x] = s;   // blocks never straddle batches
    }
}

// ---------------------------------------------------------------------------
// Phase 3: deterministic fixed-order reduction -> mean(|linking_b| / 4pi)
// ---------------------------------------------------------------------------
__global__ void __launch_bounds__(256)
finalize(const float* __restrict__ partials, float* __restrict__ out) {
    __shared__ float red[256];
    __shared__ float res;
    if (threadIdx.x == 0) res = 0.f;
    __syncthreads();
    for (int b = 0; b < BQ; ++b) {
        float s = 0.f;
        for (int k = threadIdx.x; k < BLOCKS_PER_BATCH; k += 256)
            s += partials[b * BLOCKS_PER_BATCH + k];
        red[threadIdx.x] = s;
        __syncthreads();
        for (int stride = 128; stride > 0; stride >>= 1) {
            if (threadIdx.x < stride) red[threadIdx.x] += red[threadIdx.x + stride];
            __syncthreads();
        }
        if (threadIdx.x == 0)
            res += fabsf(red[0] * (1.0f / (4.0f * 3.14159265358979323846f)));
        __syncthreads();
    }
    if (threadIdx.x == 0) out[0] = res * (1.0f / (float)BQ);
}

extern "C" void kernel_launch(void* const* d_in, const int* in_sizes, int n_in,
                              void* d_out, int out_size, void* d_ws, size_t ws_size,
                              hipStream_t stream) {
    const float* gq = (const float*)d_in[0];   // genomic_quats [4,2048,4] f32
    const float* cq = (const float*)d_in[1];   // code_quats    [4,2048,4] f32

    float* Fa       = (float*)d_ws;                       // 4*2048*16 f32 = 512 KB
    float* Fb       = Fa + (size_t)BQ * NPAD * 16;        // 512 KB
    float* partials = Fb + (size_t)BQ * NPAD * 16;        // 8192 f32 = 32 KB

    seg_features<<<(BQ * NPAD + 255) / 256, 256, 0, stream>>>(gq, cq, Fa, Fb);
    gauss_tiles<<<BQ * BLOCKS_PER_BATCH, 256, 0, stream>>>(Fa, Fb, partials);
    finalize<<<1, 256, 0, stream>>>(partials, (float*)d_out);
}